// GATModel_78821239816829
// MI455X (gfx1250) — compile-verified
//
#include <hip/hip_runtime.h>
#include <hip/hip_bf16.h>
#include <math.h>

#define NN 100000
#define NE 1600000
#define NG 2000
#define INF_ 64
#define HID 16

typedef __attribute__((ext_vector_type(2))) float v2f;
typedef __attribute__((ext_vector_type(8))) float v8f;

static __host__ int cdiv(int a, int b) { return (a + b - 1) / b; }

// ---------------------------------------------------------------------------
// Generic fill (used to init accumulators / -inf for segment max)
// ---------------------------------------------------------------------------
__global__ void fill_u32_kernel(unsigned int* __restrict__ p, unsigned int v, long n) {
    long i = (long)blockIdx.x * blockDim.x + threadIdx.x;
    long stride = (long)gridDim.x * blockDim.x;
    for (; i < n; i += stride) p[i] = v;
}

// ---------------------------------------------------------------------------
// C[n,o] = sum_k X[n,k] * W[o,k]   (X:[N,K], W:[O,K] row-major, C:[N,O])
// One wave per 16x16 output tile via V_WMMA_F32_16X16X4_F32.
// Block = 128 threads (4 waves) covering 64 rows; grid.y tiles columns.
// W tile (transposed, K x 16) staged in LDS.
// Requires N%16==0, K%4==0, O%16==0 (true for all three layers).
// ---------------------------------------------------------------------------
__global__ void gemm_xwt_wmma(const float* __restrict__ X,
                              const float* __restrict__ W,
                              float* __restrict__ C,
                              int N, int K, int O) {
    __shared__ float bt[INF_ * 16];            // K<=64 -> up to 4KB
    const int colbase = blockIdx.y * 16;
    for (int i = threadIdx.x; i < K * 16; i += blockDim.x) {
        int k = i >> 4, c = i & 15;
        bt[i] = W[(size_t)(colbase + c) * K + k];   // B^T[k][c]
    }
    __syncthreads();

    const int wave = threadIdx.x >> 5;
    const int lane = threadIdx.x & 31;
    const int rowbase = blockIdx.x * 64 + wave * 16;
    if (rowbase >= N) return;

    const int mrow = lane & 15;                // row within A tile / col within B tile
    const int koff = (lane >> 4) << 1;         // lanes 0-15: K=0,1 ; lanes 16-31: K=2,3
    const float* xrow = X + (size_t)(rowbase + mrow) * K;

    v8f acc = {};
#if __has_builtin(__builtin_amdgcn_wmma_f32_16x16x4_f32)
    for (int k0 = 0; k0 < K; k0 += 4) {
        v2f a, b;
        a.x = xrow[k0 + koff];
        a.y = xrow[k0 + koff + 1];
        b.x = bt[(k0 + koff) * 16 + mrow];
        b.y = bt[(k0 + koff + 1) * 16 + mrow];
        acc = __builtin_amdgcn_wmma_f32_16x16x4_f32(
            /*neg_a=*/false, a, /*neg_b=*/false, b,
            /*c_mod=*/(short)0, acc, /*reuse_a=*/false, /*reuse_b=*/false);
    }
#else
    // Scalar fallback matching the 16x16 f32 D layout.
    for (int r = 0; r < 8; ++r) {
        int row = rowbase + r + ((lane >> 4) << 3);
        float s = 0.f;
        for (int k = 0; k < K; ++k) s += X[(size_t)row * K + k] * bt[k * 16 + mrow];
        acc[r] = s;
    }
#endif
    // D layout: VGPR r -> M = r (lanes 0-15) / r+8 (lanes 16-31), N = lane&15
    const int rgrp = (lane >> 4) << 3;
    for (int r = 0; r < 8; ++r) {
        int row = rowbase + r + rgrp;
        C[(size_t)row * O + colbase + mrow] = acc[r];
    }
}

// ---------------------------------------------------------------------------
// el[n,h] = <feat[n,h,:], al[h,:]>, er likewise
// ---------------------------------------------------------------------------
__global__ void attn_proj_kernel(const float* __restrict__ feat,
                                 const float* __restrict__ al,
                                 const float* __restrict__ ar,
                                 float* __restrict__ el, float* __restrict__ er,
                                 int N, int H) {
    int t = blockIdx.x * blockDim.x + threadIdx.x;
    if (t >= N * H) return;
    int n = t / H, h = t - n * H;
    const float* f = feat + (size_t)n * H * HID + h * HID;
    float sl = 0.f, sr = 0.f;
    for (int i = 0; i < HID; ++i) {
        float v = f[i];
        sl = fmaf(v, al[h * HID + i], sl);
        sr = fmaf(v, ar[h * HID + i], sr);
    }
    el[t] = sl;
    er[t] = sr;
}

__device__ __forceinline__ void atomic_max_f32(float* addr, float v) {
    if (v >= 0.f) atomicMax((int*)addr, __float_as_int(v));
    else          atomicMin((unsigned int*)addr, __float_as_uint(v));
}

// ---------------------------------------------------------------------------
// Pass 1: segment max of LeakyReLU(el[src]+er[dst]) over dst
// ---------------------------------------------------------------------------
__global__ void edge_max_kernel(const int* __restrict__ src, const int* __restrict__ dst,
                                const float* __restrict__ el, const float* __restrict__ er,
                                float* __restrict__ m, int E, int H) {
    int e = blockIdx.x * blockDim.x + threadIdx.x;
    if (e >= E) return;
    int s = src[e], d = dst[e];
    for (int h = 0; h < H; ++h) {
        float sc = el[s * H + h] + er[d * H + h];
        sc = sc > 0.f ? sc : 0.2f * sc;
        atomic_max_f32(&m[d * H + h], sc);
    }
}

// ---------------------------------------------------------------------------
// Pass 2: ex = exp(score - m[dst]); stash ex; segment-sum into s[dst]
// ---------------------------------------------------------------------------
__global__ void edge_expsum_kernel(const int* __restrict__ src, const int* __restrict__ dst,
                                   const float* __restrict__ el, const float* __restrict__ er,
                                   const float* __restrict__ m,
                                   float* __restrict__ exb, float* __restrict__ ssum,
                                   int E, int H) {
    int e = blockIdx.x * blockDim.x + threadIdx.x;
    if (e >= E) return;
    int s = src[e], d = dst[e];
    for (int h = 0; h < H; ++h) {
        float sc = el[s * H + h] + er[d * H + h];
        sc = sc > 0.f ? sc : 0.2f * sc;
        float ex = __expf(sc - m[d * H + h]);
        exb[(size_t)e * H + h] = ex;
        atomicAdd(&ssum[d * H + h], ex);
    }
}

// ---------------------------------------------------------------------------
// Pass 3: out[dst,h,:] += (ex/s[dst,h]) * feat[src,h,:]   (one thread per edge*head)
// ---------------------------------------------------------------------------
__global__ void edge_aggregate_kernel(const int* __restrict__ src, const int* __restrict__ dst,
                                      const float* __restrict__ exb, const float* __restrict__ ssum,
                                      const float* __restrict__ feat, float* __restrict__ out,
                                      int E, int H) {
    int t = blockIdx.x * blockDim.x + threadIdx.x;
    if (t >= E * H) return;
    int e = t / H, h = t - e * H;
    int s = src[e], d = dst[e];
    float alpha = exb[(size_t)e * H + h] / ssum[d * H + h];
    const float4* f4 = (const float4*)(feat + (size_t)s * H * HID + h * HID);
    float* o = out + (size_t)d * H * HID + h * HID;
    for (int j = 0; j < 4; ++j) {
        float4 v = f4[j];
        atomicAdd(o + 4 * j + 0, alpha * v.x);
        atomicAdd(o + 4 * j + 1, alpha * v.y);
        atomicAdd(o + 4 * j + 2, alpha * v.z);
        atomicAdd(o + 4 * j + 3, alpha * v.w);
    }
}

// ---------------------------------------------------------------------------
// x = (relu?) (x + b[j])   with b broadcast over rows
// ---------------------------------------------------------------------------
__global__ void bias_act_kernel(float* __restrict__ x, const float* __restrict__ b,
                                int N, int D, int do_relu) {
    long t = (long)blockIdx.x * blockDim.x + threadIdx.x;
    if (t >= (long)N * D) return;
    int j = (int)(t % D);
    float v = x[t] + b[j];
    x[t] = do_relu ? fmaxf(v, 0.f) : v;
}

// ---------------------------------------------------------------------------
// Per-graph mean readout (atomic accumulate)
// ---------------------------------------------------------------------------
__global__ void readout_kernel(const float* __restrict__ y, const int* __restrict__ gid,
                               float* __restrict__ ysum, float* __restrict__ cnt, int N) {
    int n = blockIdx.x * blockDim.x + threadIdx.x;
    if (n >= N) return;
    int g = gid[n];
    for (int j = 0; j < HID; ++j) atomicAdd(&ysum[g * HID + j], y[(size_t)n * HID + j]);
    atomicAdd(&cnt[g], 1.0f);
}

// ---------------------------------------------------------------------------
// Final 3-layer MLP head, one thread per graph (2000 graphs, tiny)
// ---------------------------------------------------------------------------
__global__ void mlp_head_kernel(const float* __restrict__ ysum, const float* __restrict__ cnt,
                                const float* __restrict__ fg,
                                const float* __restrict__ l1w, const float* __restrict__ l1b,
                                const float* __restrict__ l2w, const float* __restrict__ l2b,
                                const float* __restrict__ l3w, const float* __restrict__ l3b,
                                float* __restrict__ out, int G) {
    int g = blockIdx.x * blockDim.x + threadIdx.x;
    if (g >= G) return;
    float h0[HID + 3];
    float c = fmaxf(cnt[g], 1.0f);
    for (int j = 0; j < HID; ++j) h0[j] = ysum[g * HID + j] / c;
    for (int j = 0; j < 3; ++j) h0[HID + j] = fg[g * 3 + j];
    float h1[2 * HID];
    for (int o = 0; o < 2 * HID; ++o) {
        float s = l1b[o];
        for (int k = 0; k < HID + 3; ++k) s = fmaf(l1w[o * (HID + 3) + k], h0[k], s);
        h1[o] = fmaxf(s, 0.f);
    }
    float h2[HID];
    for (int o = 0; o < HID; ++o) {
        float s = l2b[o];
        for (int k = 0; k < 2 * HID; ++k) s = fmaf(l2w[o * 2 * HID + k], h1[k], s);
        h2[o] = fmaxf(s, 0.f);
    }
    float s = l3b[0];
    for (int k = 0; k < HID; ++k) s = fmaf(l3w[k], h2[k], s);
    out[g] = s;
}

// ---------------------------------------------------------------------------
// Host orchestration
// ---------------------------------------------------------------------------
static void run_gat_layer(const float* x, int in_dim, const float* W,
                          const float* al, const float* ar, int H,
                          const int* esrc, const int* edst,
                          float* feat, float* agg,   // feat: gemm out [N,H*16]; agg: zeroed accum [N,H*16]
                          float* el, float* er, float* mb, float* sb, float* exb,
                          hipStream_t stream) {
    const int O = H * HID;
    dim3 ggrid(cdiv(NN, 64), O / 16);
    gemm_xwt_wmma<<<ggrid, 128, 0, stream>>>(x, W, feat, NN, in_dim, O);

    attn_proj_kernel<<<cdiv(NN * H, 256), 256, 0, stream>>>(feat, al, ar, el, er, NN, H);

    fill_u32_kernel<<<512, 256, 0, stream>>>((unsigned int*)mb, 0xFF800000u, (long)NN * H); // -inf
    fill_u32_kernel<<<512, 256, 0, stream>>>((unsigned int*)sb, 0u, (long)NN * H);
    fill_u32_kernel<<<2048, 256, 0, stream>>>((unsigned int*)agg, 0u, (long)NN * O);

    edge_max_kernel<<<cdiv(NE, 256), 256, 0, stream>>>(esrc, edst, el, er, mb, NE, H);
    edge_expsum_kernel<<<cdiv(NE, 256), 256, 0, stream>>>(esrc, edst, el, er, mb, exb, sb, NE, H);
    edge_aggregate_kernel<<<cdiv(NE * H, 256), 256, 0, stream>>>(esrc, edst, exb, sb, feat, agg, NE, H);
}

extern "C" void kernel_launch(void* const* d_in, const int* in_sizes, int n_in,
                              void* d_out, int out_size, void* d_ws, size_t ws_size,
                              hipStream_t stream) {
    const float* feats_node  = (const float*)d_in[0];
    const float* feats_graph = (const float*)d_in[1];
    const int*   edge_src    = (const int*)d_in[2];
    const int*   edge_dst    = (const int*)d_in[3];
    const int*   node_gid    = (const int*)d_in[4];
    const float* W1  = (const float*)d_in[5];
    const float* al1 = (const float*)d_in[6];
    const float* ar1 = (const float*)d_in[7];
    const float* W2  = (const float*)d_in[8];
    const float* al2 = (const float*)d_in[9];
    const float* ar2 = (const float*)d_in[10];
    const float* W3  = (const float*)d_in[11];
    const float* al3 = (const float*)d_in[12];
    const float* ar3 = (const float*)d_in[13];
    const float* b1  = (const float*)d_in[14];
    const float* b2  = (const float*)d_in[15];
    const float* b3  = (const float*)d_in[16];
    const float* l1w = (const float*)d_in[17];
    const float* l1b = (const float*)d_in[18];
    const float* l2w = (const float*)d_in[19];
    const float* l2b = (const float*)d_in[20];
    const float* l3w = (const float*)d_in[21];
    const float* l3b = (const float*)d_in[22];
    float* out = (float*)d_out;

    // Workspace carve-up (floats). Total ~63 MB — fits comfortably in L2 (192MB).
    float* F    = (float*)d_ws;               // [NN, 48] gemm/feat buffer
    float* G    = F  + (size_t)NN * 48;       // [NN, 48] aggregation buffer (becomes next input)
    float* el   = G  + (size_t)NN * 48;       // [NN, 3]
    float* er   = el + (size_t)NN * 3;
    float* mb   = er + (size_t)NN * 3;
    float* sb   = mb + (size_t)NN * 3;
    float* exb  = sb + (size_t)NN * 3;        // [NE, 3]
    float* ysum = exb + (size_t)NE * 3;       // [NG, 16]
    float* cnt  = ysum + (size_t)NG * HID;    // [NG]
    (void)in_sizes; (void)n_in; (void)out_size; (void)ws_size;

    // ---- Layer 1: in=64, H=3 ----  feat->F, agg->G
    run_gat_layer(feats_node, INF_, W1, al1, ar1, 3, edge_src, edge_dst,
                  F, G, el, er, mb, sb, exb, stream);
    bias_act_kernel<<<cdiv(NN * 48, 256), 256, 0, stream>>>(G, b1, NN, 48, 1);

    // ---- Layer 2: in=48, H=3 ----  x=G, feat->F, agg back into G (gemm consumed G first)
    {
        dim3 ggrid(cdiv(NN, 64), 48 / 16);
        gemm_xwt_wmma<<<ggrid, 128, 0, stream>>>(G, W2, F, NN, 48, 48);
        attn_proj_kernel<<<cdiv(NN * 3, 256), 256, 0, stream>>>(F, al2, ar2, el, er, NN, 3);
        fill_u32_kernel<<<512, 256, 0, stream>>>((unsigned int*)mb, 0xFF800000u, (long)NN * 3);
        fill_u32_kernel<<<512, 256, 0, stream>>>((unsigned int*)sb, 0u, (long)NN * 3);
        fill_u32_kernel<<<2048, 256, 0, stream>>>((unsigned int*)G, 0u, (long)NN * 48);
        edge_max_kernel<<<cdiv(NE, 256), 256, 0, stream>>>(edge_src, edge_dst, el, er, mb, NE, 3);
        edge_expsum_kernel<<<cdiv(NE, 256), 256, 0, stream>>>(edge_src, edge_dst, el, er, mb, exb, sb, NE, 3);
        edge_aggregate_kernel<<<cdiv(NE * 3, 256), 256, 0, stream>>>(edge_src, edge_dst, exb, sb, F, G, NE, 3);
        bias_act_kernel<<<cdiv(NN * 48, 256), 256, 0, stream>>>(G, b2, NN, 48, 1);
    }

    // ---- Layer 3: in=48, H=1, out=16 ----  x=G, feat->F as [NN,16], agg back into G as [NN,16]
    {
        dim3 ggrid(cdiv(NN, 64), 1);
        gemm_xwt_wmma<<<ggrid, 128, 0, stream>>>(G, W3, F, NN, 48, HID);
        attn_proj_kernel<<<cdiv(NN, 256), 256, 0, stream>>>(F, al3, ar3, el, er, NN, 1);
        fill_u32_kernel<<<512, 256, 0, stream>>>((unsigned int*)mb, 0xFF800000u, (long)NN);
        fill_u32_kernel<<<512, 256, 0, stream>>>((unsigned int*)sb, 0u, (long)NN);
        fill_u32_kernel<<<2048, 256, 0, stream>>>((unsigned int*)G, 0u, (long)NN * HID);
        edge_max_kernel<<<cdiv(NE, 256), 256, 0, stream>>>(edge_src, edge_dst, el, er, mb, NE, 1);
        edge_expsum_kernel<<<cdiv(NE, 256), 256, 0, stream>>>(edge_src, edge_dst, el, er, mb, exb, sb, NE, 1);
        edge_aggregate_kernel<<<cdiv(NE, 256), 256, 0, stream>>>(edge_src, edge_dst, exb, sb, F, G, NE, 1);
        bias_act_kernel<<<cdiv(NN * HID, 256), 256, 0, stream>>>(G, b3, NN, HID, 0);
    }

    // ---- Readout + MLP head ----
    fill_u32_kernel<<<64, 256, 0, stream>>>((unsigned int*)ysum, 0u, (long)NG * HID);
    fill_u32_kernel<<<8, 256, 0, stream>>>((unsigned int*)cnt, 0u, (long)NG);
    readout_kernel<<<cdiv(NN, 256), 256, 0, stream>>>(G, node_gid, ysum, cnt, NN);
    mlp_head_kernel<<<cdiv(NG, 128), 128, 0, stream>>>(ysum, cnt, feats_graph,
                                                       l1w, l1b, l2w, l2b, l3w, l3b,
                                                       out, NG);
}